// SynthAtt_38603166056654
// MI455X (gfx1250) — compile-verified
//
#include <hip/hip_runtime.h>
#include <hip/hip_bf16.h>

// ---------------------------------------------------------------------------
// Problem constants (reference: H=8, A=3, D=128, KD=VD=16, E=128, B=64, G=200)
// ---------------------------------------------------------------------------
#define Hh   8
#define Dd   128
#define KDim 16
#define Ee   128
#define Bb   64
#define Gg   200
#define BGr  (Bb * Gg)        // 12800 rows
#define RCh  16               // (A-1)*H route channels
#define GT   13               // ceil(G/16) g-tiles
#define QR   8                // query rows per attention block (keeps LDS < 64KB)
#define QT   25               // G / QR query tiles
#define GPAD (GT * 16 + 1)    // 209: padded LDS row stride (coprime with 64 banks)

typedef float v2f __attribute__((ext_vector_type(2)));
typedef float v8f __attribute__((ext_vector_type(8)));

// D = A(16x4,f32) * B(4x16,f32) + C(16x16,f32)  -- V_WMMA_F32_16X16X4_F32
static __device__ __forceinline__ v8f wmma4(v2f a, v2f b, v8f c) {
  return __builtin_amdgcn_wmma_f32_16x16x4_f32(false, a, false, b, (short)0, c,
                                               false, false);
}

// ---------------------------------------------------------------------------
// Kernel 1: Q/K/V projections.  grid = (200, 8) blocks of 128 threads.
// Wave w of block x handles row-tile (4x + w) of head blockIdx.y.
// Per tile: 32 K-chunks x 3 outputs = 96 WMMAs.
// ---------------------------------------------------------------------------
__global__ __launch_bounds__(128) void qkv_kernel(
    const float* __restrict__ h_em, const float* __restrict__ Wq,
    const float* __restrict__ Wk, const float* __restrict__ Wv,
    float* __restrict__ Q, float* __restrict__ K, float* __restrict__ V) {
  const int lane  = threadIdx.x & 31;
  const int wave  = threadIdx.x >> 5;
  const int tile  = blockIdx.x * 4 + wave;   // 0..799, exact
  const int h     = blockIdx.y;
  const int n     = lane & 15;
  const int khalf = lane >> 4;

  const float* Arow = h_em + (size_t)(tile * 16 + n) * Dd;   // A row m = n
  const float* wq = Wq + (size_t)h * Dd * KDim + n;
  const float* wk = Wk + (size_t)h * Dd * KDim + n;
  const float* wv = Wv + (size_t)h * Dd * KDim + n;

  v8f cq = {}; v8f ck = {}; v8f cv = {};
  #pragma unroll 4
  for (int kk = 0; kk < Dd; kk += 4) {
    const int k0 = kk + khalf * 2;
    v2f a;  a.x  = Arow[k0];             a.y  = Arow[k0 + 1];
    v2f bq; bq.x = wq[k0 * KDim];        bq.y = wq[(k0 + 1) * KDim];
    v2f bk; bk.x = wk[k0 * KDim];        bk.y = wk[(k0 + 1) * KDim];
    v2f bv; bv.x = wv[k0 * KDim];        bv.y = wv[(k0 + 1) * KDim];
    cq = wmma4(a, bq, cq);
    ck = wmma4(a, bk, ck);
    cv = wmma4(a, bv, cv);
  }
  // C rows: M = r + 8*khalf
  const size_t base = ((size_t)h * BGr + tile * 16 + 8 * khalf) * KDim + n;
  #pragma unroll
  for (int r = 0; r < 8; ++r) {
    Q[base + (size_t)r * KDim] = cq[r];
    K[base + (size_t)r * KDim] = ck[r];
    V[base + (size_t)r * KDim] = cv[r];
  }
}

// ---------------------------------------------------------------------------
// Kernel 2: fused  qk(WMMA) -> channel-MLP (+route_attn passthrough copy)
//                 -> softmax -> P@V (WMMA).
// grid = B * 25 blocks of 128 threads (4 waves; wave w owns heads 2w, 2w+1).
// LDS: scores S[8][8][209] + MLP params + row sums  (~56 KB, CU-mode safe).
// ---------------------------------------------------------------------------
__global__ __launch_bounds__(128) void attn_kernel(
    const float* __restrict__ route,
    const float* __restrict__ Q, const float* __restrict__ K,
    const float* __restrict__ V,
    const float* __restrict__ W1, const float* __restrict__ b1,
    const float* __restrict__ W2, const float* __restrict__ b2,
    float* __restrict__ heads, float* __restrict__ route_out) {
  __shared__ float S[Hh][QR][GPAD];
  __shared__ float W1s[16][24];
  __shared__ float W2s[8][16];
  __shared__ float b1s[16];
  __shared__ float b2s[8];
  __shared__ float Rs[Hh][QR];

  const int tid   = threadIdx.x;
  const int lane  = tid & 31;
  const int wave  = tid >> 5;
  const int b     = blockIdx.x / QT;
  const int qt    = blockIdx.x % QT;
  const int q0    = qt * QR;
  const int n     = lane & 15;
  const int khalf = lane >> 4;
  const int h0    = wave * 2;
  const int h1    = wave * 2 + 1;

  for (int i = tid; i < 16 * 24; i += 128) ((float*)W1s)[i] = W1[i];
  if (tid < 128) ((float*)W2s)[tid] = W2[tid];
  if (tid < 16)  b1s[tid] = b1[tid];
  if (tid < 8)   b2s[tid] = b2[tid];

  // Q rows in WMMA A layout (KD=16 -> 4 chunks of k=4), clamped at tail.
  int qrow = q0 + n; if (qrow >= Gg) qrow = Gg - 1;
  const float* q0p = Q + ((size_t)h0 * BGr + (size_t)b * Gg + qrow) * KDim;
  const float* q1p = Q + ((size_t)h1 * BGr + (size_t)b * Gg + qrow) * KDim;
  v2f qa0[4], qa1[4];
  #pragma unroll
  for (int c = 0; c < 4; ++c) {
    const int k0 = c * 4 + khalf * 2;
    qa0[c].x = q0p[k0]; qa0[c].y = q0p[k0 + 1];
    qa1[c].x = q1p[k0]; qa1[c].y = q1p[k0 + 1];
  }
  __syncthreads();

  const size_t kvb0 = (size_t)h0 * BGr + (size_t)b * Gg;
  const size_t kvb1 = (size_t)h1 * BGr + (size_t)b * Gg;
  const size_t rstride = (size_t)Bb * Gg * Gg;   // per-channel route stride

  for (int gtile = 0; gtile < GT; ++gtile) {
    const int g0 = gtile * 16;
    int gcol = g0 + n; if (gcol >= Gg) gcol = Gg - 1;

    // ---- qk tile for 2 heads: 4+4 WMMAs ----
    v8f c0 = {}; v8f c1 = {};
    #pragma unroll
    for (int c = 0; c < 4; ++c) {
      const int k0 = c * 4 + khalf * 2;
      const float* kp0 = K + (kvb0 + gcol) * KDim + k0;
      const float* kp1 = K + (kvb1 + gcol) * KDim + k0;
      v2f kb0; kb0.x = kp0[0]; kb0.y = kp0[1];
      v2f kb1; kb1.x = kp1[0]; kb1.y = kp1[1];
      c0 = wmma4(qa0[c], kb0, c0);
      c1 = wmma4(qa1[c], kb1, c1);
    }
    if (khalf == 0) {                      // rows M=0..7 live in lanes 0-15
      #pragma unroll
      for (int r = 0; r < 8; ++r) {
        S[h0][r][g0 + n] = c0[r];
        S[h1][r][g0 + n] = c1[r];
      }
    }
    __syncthreads();

    // ---- per-position channel MLP (24->16->8) + fused route_attn copy ----
    {
      const int ql = tid >> 4;             // 0..7
      const int gl = tid & 15;             // 0..15
      const int q  = q0 + ql;
      const int g  = g0 + gl;
      float qk8[8];
      #pragma unroll
      for (int hh = 0; hh < 8; ++hh) qk8[hh] = S[hh][ql][g0 + gl];

      const int qc = q < Gg ? q : Gg - 1;
      const int gc = g < Gg ? g : Gg - 1;
      const size_t rbase = ((size_t)b * Gg + qc) * Gg + gc;
      float rv[RCh];
      #pragma unroll
      for (int c = 0; c < RCh; ++c)
        rv[c] = __builtin_nontemporal_load(route + (size_t)c * rstride + rbase);

      float hid[16];
      #pragma unroll
      for (int j = 0; j < 16; ++j) {
        float acc = b1s[j];
        #pragma unroll
        for (int hh = 0; hh < 8; ++hh) acc += qk8[hh] * W1s[j][hh];
        #pragma unroll
        for (int c = 0; c < RCh; ++c)   acc += rv[c] * W1s[j][8 + c];
        hid[j] = acc > 0.f ? acc : 0.f;
      }
      const float NEG_INF = -__builtin_inff();
      #pragma unroll
      for (int hh = 0; hh < 8; ++hh) {
        float acc = b2s[hh];
        #pragma unroll
        for (int j = 0; j < 16; ++j) acc += hid[j] * W2s[hh][j];
        S[hh][ql][g0 + gl] = (g < Gg) ? acc : NEG_INF;   // in-place, thread-private
      }
      if (q < Gg && g < Gg) {              // pass route_attn through to output
        #pragma unroll
        for (int c = 0; c < RCh; ++c)
          __builtin_nontemporal_store(rv[c], route_out + (size_t)c * rstride + rbase);
      }
    }
    __syncthreads();
  }

  // ---- softmax over g for 64 (head,row) pairs; exp() stored in place ----
  if (tid < 64) {
    const int hh = tid >> 3;
    const int ql = tid & 7;
    float m = -__builtin_inff();
    for (int g = 0; g < GT * 16; ++g) { float x = S[hh][ql][g]; m = x > m ? x : m; }
    float s = 0.f;
    for (int g = 0; g < GT * 16; ++g) {
      float e = __expf(S[hh][ql][g] - m);
      S[hh][ql][g] = e;                    // padded cols: exp(-inf)=0
      s += e;
    }
    Rs[hh][ql] = s;
  }
  __syncthreads();

  // ---- P~ @ V via WMMA (13 tiles x 4 chunks x 2 heads) ----
  v8f o0 = {}; v8f o1 = {};
  const int mq = n & 7;                    // duplicate A rows 8..15 (unused)
  for (int gtile = 0; gtile < GT; ++gtile) {
    const int g0 = gtile * 16;
    #pragma unroll
    for (int c = 0; c < 4; ++c) {
      const int k0 = c * 4 + khalf * 2;
      v2f pa0, pa1;
      pa0.x = S[h0][mq][g0 + k0]; pa0.y = S[h0][mq][g0 + k0 + 1];
      pa1.x = S[h1][mq][g0 + k0]; pa1.y = S[h1][mq][g0 + k0 + 1];
      const int gr  = g0 + k0;
      const int g_a = gr < Gg ? gr : Gg - 1;             // P~=0 there anyway
      const int g_b = (gr + 1) < Gg ? (gr + 1) : Gg - 1;
      v2f vb0, vb1;
      vb0.x = V[(kvb0 + g_a) * KDim + n]; vb0.y = V[(kvb0 + g_b) * KDim + n];
      vb1.x = V[(kvb1 + g_a) * KDim + n]; vb1.y = V[(kvb1 + g_b) * KDim + n];
      o0 = wmma4(pa0, vb0, o0);
      o1 = wmma4(pa1, vb1, o1);
    }
  }
  if (khalf == 0) {
    #pragma unroll
    for (int r = 0; r < 8; ++r) {
      const int q = q0 + r;
      if (q < Gg) {
        heads[((size_t)h0 * BGr + (size_t)b * Gg + q) * KDim + n] = o0[r] / Rs[h0][r];
        heads[((size_t)h1 * BGr + (size_t)b * Gg + q) * KDim + n] = o1[r] / Rs[h1][r];
      }
    }
  }
}

// ---------------------------------------------------------------------------
// Kernel 3: out[b,g,e] = sum_{h,v} heads[h,bg,v] * W_out[h,v,e]
// grid = 800 blocks of 256 threads; wave w owns e-tile w (8 x 16 = 128 = E).
// 32 WMMAs per 16x16 output tile.
// ---------------------------------------------------------------------------
__global__ __launch_bounds__(256) void outproj_kernel(
    const float* __restrict__ heads, const float* __restrict__ Wout,
    float* __restrict__ out) {
  const int lane  = threadIdx.x & 31;
  const int et    = threadIdx.x >> 5;     // e-tile 0..7
  const int tile  = blockIdx.x;           // row tile 0..799, exact
  const int n     = lane & 15;
  const int khalf = lane >> 4;

  v8f acc = {};
  for (int h = 0; h < Hh; ++h) {
    const float* hb = heads + ((size_t)h * BGr + tile * 16 + n) * KDim;
    const float* wb = Wout + (size_t)h * KDim * Ee + et * 16 + n;
    #pragma unroll
    for (int c = 0; c < 4; ++c) {
      const int k0 = c * 4 + khalf * 2;
      v2f a;  a.x  = hb[k0];          a.y  = hb[k0 + 1];
      v2f bm; bm.x = wb[k0 * Ee];     bm.y = wb[(k0 + 1) * Ee];
      acc = wmma4(a, bm, acc);
    }
  }
  const size_t obase = ((size_t)tile * 16 + 8 * khalf) * Ee + et * 16 + n;
  #pragma unroll
  for (int r = 0; r < 8; ++r)
    __builtin_nontemporal_store(acc[r], &out[obase + (size_t)r * Ee]);
}

// ---------------------------------------------------------------------------
// Host launcher
// ---------------------------------------------------------------------------
extern "C" void kernel_launch(void* const* d_in, const int* in_sizes, int n_in,
                              void* d_out, int out_size, void* d_ws, size_t ws_size,
                              hipStream_t stream) {
  const float* h_em  = (const float*)d_in[0];   // [B,G,D]
  const float* route = (const float*)d_in[1];   // [16,B,G,G]
  const float* Wq    = (const float*)d_in[2];   // [H,D,16]
  const float* Wk    = (const float*)d_in[3];
  const float* Wv    = (const float*)d_in[4];
  const float* W1    = (const float*)d_in[5];   // [16,24]
  const float* b1    = (const float*)d_in[6];   // [16]
  const float* W2    = (const float*)d_in[7];   // [8,16]
  const float* b2    = (const float*)d_in[8];   // [8]
  const float* Wout  = (const float*)d_in[9];   // [H,16,E]
  (void)in_sizes; (void)n_in; (void)out_size; (void)ws_size;

  float* out       = (float*)d_out;                         // [B,G,E]
  float* route_out = out + (size_t)Bb * Gg * Ee;            // [16,B,G,G]

  const size_t seg = (size_t)Hh * BGr * KDim;               // 1,638,400 floats
  float* Q     = (float*)d_ws;
  float* K     = Q + seg;
  float* V     = K + seg;
  float* heads = V + seg;                                   // total ~26.2 MB

  qkv_kernel<<<dim3(200, Hh), 128, 0, stream>>>(h_em, Wq, Wk, Wv, Q, K, V);
  attn_kernel<<<dim3(Bb * QT), 128, 0, stream>>>(route, Q, K, V, W1, b1, W2, b2,
                                                 heads, route_out);
  outproj_kernel<<<dim3(BGr / 16), 256, 0, stream>>>(heads, Wout, out);
}